// PatchAggregator_46162308497424
// MI455X (gfx1250) — compile-verified
//
#include <hip/hip_runtime.h>

#define PS 64
#define VECS_PER_PATCH (PS * PS / 4)   // 1024 float4 per patch
#define FILL_VALUE (-10.0f)

typedef float        v4f __attribute__((ext_vector_type(4)));
typedef unsigned int v4u __attribute__((ext_vector_type(4)));

// ---------------------------------------------------------------------------
// Kernel 1: zero the sum buffer (d_out) and the count buffer (d_ws).
// Regular-temporal 16B stores: zeroed accumulators should stay L2-resident
// for the atomic phase.
// ---------------------------------------------------------------------------
__global__ void pa_init_kernel(v4f* __restrict__ sum4,
                               v4u* __restrict__ cnt4,
                               int n4) {
    int i = blockIdx.x * blockDim.x + threadIdx.x;
    if (i >= n4) return;
    sum4[i] = (v4f)(0.0f);
    cnt4[i] = (v4u)(0u);
}

// ---------------------------------------------------------------------------
// Kernel 2: scatter-add. One thread per float4 of patch data.
// Thread i -> patch p = i >> 10 (uniform across a wave32: 1024 % 32 == 0),
// so coords / batch math are scalarized via readfirstlane -> s_load / SALU.
// Patch stream is read once -> non-temporal load (keeps accumulators in L2).
// Sums: 4x global_atomic_add_f32 (non-returning).
// Counts: packed u64 atomic (+1,+1) when 8B-aligned, else 4x u32 atomics;
// the branch is scalar-uniform (cw is uniform per wave).
// ---------------------------------------------------------------------------
__global__ void pa_scatter_kernel(const v4f* __restrict__ patches,
                                  const int* __restrict__ coords,
                                  float* __restrict__ sums,
                                  unsigned int* __restrict__ counts,
                                  const int* __restrict__ Hp,
                                  const int* __restrict__ Wp,
                                  int nvec,        // total float4 count = B*K*1024
                                  int npatch,      // B*K
                                  int out_elems) { // B*H*W
    int i = blockIdx.x * blockDim.x + threadIdx.x;
    if (i >= nvec) return;

    // Uniform per wave: patch id and everything derived from it.
    int p = __builtin_amdgcn_readfirstlane(i >> 10);   // i / VECS_PER_PATCH
    int v = i & (VECS_PER_PATCH - 1);

    int H  = Hp[0];                 // scalar (uniform) loads
    int W  = Wp[0];
    int HW = H * W;
    int B  = out_elems / HW;
    int K  = npatch / B;
    int b  = p / K;                 // SALU divide (p uniform)

    int ch = coords[2 * p + 0];     // s_load: scalar base + scalar offset
    int cw = coords[2 * p + 1];

    // Within-patch position: 16 float4 per row.
    int row = v >> 4;
    int col = (v & 15) << 2;

    // Prefetch the patch stream a little ahead (gfx1250 global_prefetch_b8).
    __builtin_prefetch((const char*)(patches + i) + 4096, 0, 1);

    // Read-once streaming data: non-temporal b128 load.
    v4f val = __builtin_nontemporal_load(patches + i);

    int base = b * HW + (ch + row) * W + (cw + col);
    float* s = sums + base;
    atomicAdd(s + 0, val.x);
    atomicAdd(s + 1, val.y);
    atomicAdd(s + 2, val.z);
    atomicAdd(s + 3, val.w);

    // base parity == cw parity when W and b*HW are even (W=2048 here).
    if (((W & 1) == 0) && ((cw & 1) == 0)) {
        unsigned long long* c8 = (unsigned long long*)(counts + base);
        const unsigned long long one_one = 0x0000000100000001ull;
        atomicAdd(c8 + 0, one_one);   // +1 to counts[base],   counts[base+1]
        atomicAdd(c8 + 1, one_one);   // +1 to counts[base+2], counts[base+3]
    } else {
        unsigned int* c = counts + base;
        atomicAdd(c + 0, 1u);
        atomicAdd(c + 1, 1u);
        atomicAdd(c + 2, 1u);
        atomicAdd(c + 3, 1u);
    }
}

// ---------------------------------------------------------------------------
// Kernel 3: normalize. Integer count >= 1  <=>  count > 1e-6, and then
// max(count, 1e-6) == count, so out = count ? sum/count : FILL_VALUE.
// Sums/counts are last-use reads, output is write-once: all non-temporal.
// ---------------------------------------------------------------------------
__global__ void pa_finalize_kernel(v4f* __restrict__ out4,
                                   const v4u* __restrict__ cnt4,
                                   int n4) {
    int i = blockIdx.x * blockDim.x + threadIdx.x;
    if (i >= n4) return;
    v4f s = __builtin_nontemporal_load(out4 + i);
    v4u c = __builtin_nontemporal_load(cnt4 + i);
    v4f r;
    r.x = c.x ? s.x / (float)c.x : FILL_VALUE;
    r.y = c.y ? s.y / (float)c.y : FILL_VALUE;
    r.z = c.z ? s.z / (float)c.z : FILL_VALUE;
    r.w = c.w ? s.w / (float)c.w : FILL_VALUE;
    __builtin_nontemporal_store(r, out4 + i);
}

extern "C" void kernel_launch(void* const* d_in, const int* in_sizes, int n_in,
                              void* d_out, int out_size, void* d_ws, size_t ws_size,
                              hipStream_t stream) {
    const v4f* patches = (const v4f*)d_in[0];             // [B,K,1,64,64] f32
    const int* coords  = (const int*)d_in[1];             // [B,K,2] i32
    const int* Hp      = (const int*)d_in[2];             // scalar H
    const int* Wp      = (const int*)d_in[3];             // scalar W

    float*        sums   = (float*)d_out;                 // B*H*W f32 accumulators -> final output
    unsigned int* counts = (unsigned int*)d_ws;           // B*H*W u32 hit counts

    const int n_logits = in_sizes[0];                     // B*K*64*64
    const int npatch   = in_sizes[1] / 2;                 // B*K
    const int nvec     = n_logits / 4;                    // float4 count
    const int n4       = out_size / 4;                    // H*W divisible by 4

    const int BLK = 256;                                  // 8 wave32 per block

    pa_init_kernel<<<(n4 + BLK - 1) / BLK, BLK, 0, stream>>>(
        (v4f*)sums, (v4u*)counts, n4);

    pa_scatter_kernel<<<(nvec + BLK - 1) / BLK, BLK, 0, stream>>>(
        patches, coords, sums, counts, Hp, Wp, nvec, npatch, out_size);

    pa_finalize_kernel<<<(n4 + BLK - 1) / BLK, BLK, 0, stream>>>(
        (v4f*)sums, (const v4u*)counts, n4);
}